// DynamicDWConv_14173392076921
// MI455X (gfx1250) — compile-verified
//
#include <hip/hip_runtime.h>
#include <math.h>

// Problem constants (from the reference)
#define BATCH 32
#define CH    256
#define HGT   64
#define WID   64
#define HID   64        // CH / 4
#define KKT   2304      // CH * 3 * 3
#define HW    4096      // HGT * WID

typedef float v2f __attribute__((ext_vector_type(2)));
typedef float v8f __attribute__((ext_vector_type(8)));

// ---------------------------------------------------------------------------
// Kernel 1: global average pool over each (b, c) plane.  Pure streaming read:
// 128 MiB at 23.3 TB/s ~ 5.7 us.  One block per plane, float4 loads, LDS
// tree reduction.
// ---------------------------------------------------------------------------
__global__ __launch_bounds__(256) void pool_kernel(const float* __restrict__ x,
                                                   float* __restrict__ pooled) {
    __shared__ float red[256];
    const int bc  = blockIdx.x;                 // 0 .. BATCH*CH-1
    const int tid = threadIdx.x;
    const float4* xp = reinterpret_cast<const float4*>(x + (size_t)bc * HW);
    float s = 0.0f;
#pragma unroll
    for (int i = 0; i < 4; ++i) {
        float4 v = xp[tid + 256 * i];
        s += v.x + v.y + v.z + v.w;
    }
    red[tid] = s;
    __syncthreads();
#pragma unroll
    for (int off = 128; off > 0; off >>= 1) {
        if (tid < off) red[tid] += red[tid + off];
        __syncthreads();
    }
    if (tid == 0) pooled[bc] = red[0] * (1.0f / (float)HW);
}

// ---------------------------------------------------------------------------
// Kernel 2: y = sigmoid(BN(pooled @ w1^T))  -> [32, 64].  2048 dot-256
// products: completely negligible, one block is plenty.
// ---------------------------------------------------------------------------
__global__ __launch_bounds__(256) void se_kernel(const float* __restrict__ pooled,
                                                 const float* __restrict__ w1,
                                                 const float* __restrict__ gamma,
                                                 const float* __restrict__ beta,
                                                 const float* __restrict__ mean,
                                                 const float* __restrict__ var,
                                                 float* __restrict__ y) {
    const int tid = threadIdx.x;
    for (int o = tid; o < BATCH * HID; o += 256) {
        const int b = o / HID;
        const int r = o % HID;
        const float* pr = pooled + b * CH;
        const float* wr = w1 + r * CH;
        float acc = 0.0f;
#pragma unroll 4
        for (int k = 0; k < CH; k += 4) {
            acc += pr[k + 0] * wr[k + 0] + pr[k + 1] * wr[k + 1] +
                   pr[k + 2] * wr[k + 2] + pr[k + 3] * wr[k + 3];
        }
        float t = (acc - mean[r]) * rsqrtf(var[r] + 1e-5f) * gamma[r] + beta[r];
        y[o] = 1.0f / (1.0f + expf(-t));
    }
}

// ---------------------------------------------------------------------------
// Kernel 3: dyn = y @ w2^T + b2  -> [32, 2304], via V_WMMA_F32_16X16X4_F32
// (fp32 A/B, fp32 accumulate: matches the fp32 reference exactly).
// Tiling: M=batch (2 tiles of 16), N=k (144 tiles of 16), K=hid (16 steps
// of 4).  One wave per 16x16 D tile; 8 waves/block; 36 blocks = 288 tiles.
// EXEC is all ones: every thread participates, no divergence around WMMA.
//
// f32 A (16x4) fragment per lane:  lanes 0-15 hold {K=0,K=1} of row M=lane,
// lanes 16-31 hold {K=2,K=3} of row M=lane-16.  B (4x16) is the transposed
// analog: lane holds {K=kk,K=kk+1} of column N=lane&15.  w2 is [2304,64]
// row-major, so the B fragment is a contiguous float2 -> coalesced.
// D (16x16 f32): VGPR v holds row (v + 8*(lane>=16)), col = lane&15.
// ---------------------------------------------------------------------------
__global__ __launch_bounds__(256) void dyn_kernel(const float* __restrict__ y,
                                                  const float* __restrict__ w2,
                                                  const float* __restrict__ b2,
                                                  float* __restrict__ dyn) {
    const int wave = threadIdx.x >> 5;
    const int lane = threadIdx.x & 31;
    const int tile = blockIdx.x * 8 + wave;    // 0 .. 287
    const int tm   = tile / 144;               // batch-tile (0..1)
    const int tn   = tile % 144;               // k-tile (0..143)

    const int half = lane >> 4;                // 0: lanes 0-15, 1: lanes 16-31
    const int kk   = half * 2;                 // K offset within 4-wide step
    const int arow = tm * 16 + (lane & 15);    // batch row carried by this lane (A)
    const int bcol = tn * 16 + (lane & 15);    // output column carried by this lane (B)

    const float* yA = y  + (size_t)arow * HID;
    const float* wB = w2 + (size_t)bcol * HID;

    v8f acc = {};
#pragma unroll
    for (int k0 = 0; k0 < HID; k0 += 4) {
        v2f a, b;
        a.x = yA[k0 + kk];
        a.y = yA[k0 + kk + 1];
        b.x = wB[k0 + kk];
        b.y = wB[k0 + kk + 1];
        // 8 args: (neg_a, A, neg_b, B, c_mod, C, reuse_a, reuse_b)
        acc = __builtin_amdgcn_wmma_f32_16x16x4_f32(
            false, a, false, b, (short)0, acc, false, false);
    }

    const int col   = tn * 16 + (lane & 15);
    const int rbase = tm * 16 + (half << 3);
    const float bb  = b2[col];
#pragma unroll
    for (int v = 0; v < 8; ++v) {
        dyn[(size_t)(rbase + v) * KKT + col] = acc[v] + bb;
    }
}

// ---------------------------------------------------------------------------
// Kernel 4: per-(b,c) depthwise 3x3 conv with the dynamic kernel + bias.
// One block per 64x64 plane.  The 16 KiB plane is staged into LDS with
// GLOBAL_LOAD_ASYNC_TO_LDS_B128 (CDNA5 async copy: memory -> LDS directly,
// no VGPR round-trip, tracked on ASYNCcnt), then 3x3 depthwise out of LDS
// with zero-padding bounds checks, fused bias, 32-bit coalesced stores.
// dyn flat index for (b,c): (b*CH + c)*9 + j == bc*9 + j.
// The input re-read should largely hit the 192 MB L2 (x fits after pass 1).
// ---------------------------------------------------------------------------
__global__ __launch_bounds__(256) void conv_kernel(const float* __restrict__ x,
                                                   const float* __restrict__ dyn,
                                                   const float* __restrict__ bias,
                                                   float* __restrict__ out) {
    __shared__ float tile[HW];                  // 16 KiB plane
    __shared__ float skw[9];                    // broadcast dyn 3x3 kernel
    const int bc  = blockIdx.x;
    const int cch = bc & (CH - 1);
    const int tid = threadIdx.x;

    // --- async global -> LDS staging (GVS mode: SGPR base + VGPR offset) ---
    // VDST VGPR carries the workgroup-relative LDS byte address = low 32 bits
    // of the generic pointer (aperture bits live in [63:32] per the ISA).
    const float* src = x + (size_t)bc * HW;
    const unsigned lds_base = (unsigned)(uintptr_t)(&tile[0]);
#pragma unroll
    for (int i = 0; i < 4; ++i) {
        const unsigned goff = (unsigned)(tid + 256 * i) * 16u;  // byte offset
        const unsigned ldst = lds_base + goff;
        asm volatile("global_load_async_to_lds_b128 %0, %1, %2"
                     :: "v"(ldst), "v"(goff), "s"(src)
                     : "memory");
    }

    // stage the 9 dynamic weights once per block (not once per thread)
    if (tid < 9) skw[tid] = dyn[(size_t)bc * 9 + tid];

    // wait for this wave's async copies, then workgroup-wide barrier
    asm volatile("s_wait_asynccnt 0x0" ::: "memory");
    __syncthreads();

    float wk[9];
#pragma unroll
    for (int j = 0; j < 9; ++j) wk[j] = skw[j];
    const float bb = bias[cch];

    const int xq = tid & 63;                    // column (lanes -> consecutive x)
    const int y0 = tid >> 6;                    // 0..3
    float* op = out + (size_t)bc * HW;
#pragma unroll
    for (int i = 0; i < 16; ++i) {
        const int yy = y0 + 4 * i;
        float acc = 0.0f;
#pragma unroll
        for (int dy = -1; dy <= 1; ++dy) {
            const int sy = yy + dy;
            if ((unsigned)sy >= (unsigned)HGT) continue;
#pragma unroll
            for (int dx = -1; dx <= 1; ++dx) {
                const int sx = xq + dx;
                if ((unsigned)sx >= (unsigned)WID) continue;
                acc += wk[(dy + 1) * 3 + (dx + 1)] * tile[sy * WID + sx];
            }
        }
        op[yy * WID + xq] = acc + bb;
    }
}

// ---------------------------------------------------------------------------
// Launch.  Workspace layout (floats): pooled[8192] | y[2048] | dyn[73728]
// = 335,872 bytes total.
// ---------------------------------------------------------------------------
extern "C" void kernel_launch(void* const* d_in, const int* in_sizes, int n_in,
                              void* d_out, int out_size, void* d_ws, size_t ws_size,
                              hipStream_t stream) {
    (void)in_sizes; (void)n_in; (void)out_size; (void)ws_size;

    const float* x     = (const float*)d_in[0];
    const float* w1    = (const float*)d_in[1];
    const float* gamma = (const float*)d_in[2];
    const float* beta  = (const float*)d_in[3];
    const float* mean  = (const float*)d_in[4];
    const float* var   = (const float*)d_in[5];
    const float* w2    = (const float*)d_in[6];
    const float* b2    = (const float*)d_in[7];
    const float* bias  = (const float*)d_in[8];
    float* out = (float*)d_out;

    float* pooled = (float*)d_ws;
    float* y      = pooled + BATCH * CH;
    float* dyn    = y + BATCH * HID;

    pool_kernel<<<BATCH * CH, 256, 0, stream>>>(x, pooled);
    se_kernel<<<1, 256, 0, stream>>>(pooled, w1, gamma, beta, mean, var, y);
    dyn_kernel<<<36, 256, 0, stream>>>(y, w2, b2, dyn);
    conv_kernel<<<BATCH * CH, 256, 0, stream>>>(x, dyn, bias, out);
}